// GraphConvolutionBlock_20366734917864
// MI455X (gfx1250) — compile-verified
//
#include <hip/hip_runtime.h>
#include <cstdint>

#define BN_EPS 1e-5f

typedef __attribute__((ext_vector_type(2))) float v2f;
typedef __attribute__((ext_vector_type(8))) float v8f;

#if __has_builtin(__builtin_amdgcn_global_load_async_to_lds_b128) && \
    __has_builtin(__builtin_amdgcn_s_wait_asynccnt)
#define HAVE_ASYNC_LDS 1
// builtin parameter type per hipcc diagnostic:
//   __attribute__((__vector_size__(4 * sizeof(int)))) int *
typedef int async_v4i __attribute__((vector_size(16)));
#else
#define HAVE_ASYNC_LDS 0
#endif

__device__ __forceinline__ float elu1(float x) {
    return x > 0.f ? x : (expf(x) - 1.f);
}

// ---------------------------------------------------------------------------
// GEMM: H[N x 128] = act(A)[N x 128] @ W[128 x 128]
//   - W staged once per workgroup in LDS (64KB) via async global->LDS copy
//     (GLOBAL_LOAD_ASYNC_TO_LDS_B128, ASYNCcnt) when available
//   - 8 waves / block, each wave owns a 16-row x 128-col strip
//   - FUSE: apply y = elu(a*scale[c] + shift[c]) to A elements on load
// WMMA: V_WMMA_F32_16X16X4_F32, 32 K-steps of 4, 8 column tiles per wave.
// ---------------------------------------------------------------------------
template <bool FUSE>
__global__ __launch_bounds__(256) void gemm_nn_128(
    const float* __restrict__ A, const float* __restrict__ W,
    const float* __restrict__ scale, const float* __restrict__ shift,
    float* __restrict__ H, int N)
{
    __shared__ float Wl[128 * 128];  // 64 KB of the 320 KB WGP LDS

#if HAVE_ASYNC_LDS
    {
        // 4096 float4 elements, 256 threads -> 16 async 16B copies per thread,
        // written straight to LDS without a VGPR round-trip.
#pragma unroll
        for (int j = 0; j < 16; ++j) {
            const int idx = threadIdx.x + j * 256;  // float4 index
            __builtin_amdgcn_global_load_async_to_lds_b128(
                (async_v4i*)(W + idx * 4), (async_v4i*)(Wl + idx * 4), 0, 0);
        }
        __builtin_amdgcn_s_wait_asynccnt(0);
    }
#else
    {
        const float4* Wv = (const float4*)W;
        float4* Wlv = (float4*)Wl;
#pragma unroll
        for (int j = 0; j < 16; ++j)
            Wlv[threadIdx.x + j * 256] = Wv[threadIdx.x + j * 256];
    }
#endif
    __syncthreads();

    const int wave = threadIdx.x >> 5;
    const int lane = threadIdx.x & 31;
    const int ln16 = lane & 15;
    const int lh   = lane >> 4;  // 0 or 1
    const int rowBase = blockIdx.x * 128 + wave * 16;

    int rowA = rowBase + ln16;           // clamp so EXEC stays all-ones for WMMA
    if (rowA > N - 1) rowA = N - 1;
    const float* Arow = A + (size_t)rowA * 128;

    v8f acc[8] = {};

    for (int kk = 0; kk < 32; ++kk) {
        const int k0 = kk * 4 + 2 * lh;  // this lane's K-pair base
        float2 av = *(const float2*)(Arow + k0);
        if (FUSE) {
            float2 sc = *(const float2*)(scale + k0);
            float2 sh = *(const float2*)(shift + k0);
            av.x = elu1(av.x * sc.x + sh.x);
            av.y = elu1(av.y * sc.y + sh.y);
        }
        v2f afrag;
        afrag[0] = av.x;
        afrag[1] = av.y;
#pragma unroll
        for (int ct = 0; ct < 8; ++ct) {
            v2f bfrag;
            bfrag[0] = Wl[(k0 + 0) * 128 + ct * 16 + ln16];
            bfrag[1] = Wl[(k0 + 1) * 128 + ct * 16 + ln16];
            acc[ct] = __builtin_amdgcn_wmma_f32_16x16x4_f32(
                false, afrag, false, bfrag, (short)0, acc[ct], false, false);
        }
    }

    // C/D layout: lane L, VGPR v -> row rowBase + v + 8*(L/16), col = L%16
#pragma unroll
    for (int ct = 0; ct < 8; ++ct) {
        const int col = ct * 16 + ln16;
#pragma unroll
        for (int v = 0; v < 8; ++v) {
            const int r = rowBase + v + 8 * lh;
            if (r < N) H[(size_t)r * 128 + col] = acc[ct][v];
        }
    }
}

// ---------------------------------------------------------------------------
// Degree / normalization helpers
// ---------------------------------------------------------------------------
__global__ __launch_bounds__(256) void zero_f32(float* p, int n) {
    int i = blockIdx.x * 256 + threadIdx.x;
    if (i < n) p[i] = 0.f;
}

__global__ __launch_bounds__(256) void deg_count(const long long* __restrict__ dst,
                                                 float* __restrict__ deg, int E) {
    int i = blockIdx.x * 256 + threadIdx.x;
    if (i < E) atomicAdd(&deg[dst[i]], 1.0f);
}

__global__ __launch_bounds__(256) void finish_dinv(float* __restrict__ deg, int n) {
    int i = blockIdx.x * 256 + threadIdx.x;
    if (i < n) deg[i] = rsqrtf(deg[i] + 1.0f);  // +1 = self loop
}

// out[i][c] = h[i][c] * dinv[i]^2 + b[c]  (self-loop term + bias)
__global__ __launch_bounds__(256) void init_out(const float* __restrict__ h,
                                                const float* __restrict__ dinv,
                                                const float* __restrict__ b,
                                                float* __restrict__ out, int n128) {
    int i = blockIdx.x * 256 + threadIdx.x;
    if (i < n128) {
        int r = i >> 7, c = i & 127;
        float di = dinv[r];
        out[i] = h[i] * di * di + b[c];
    }
}

// one wave per edge: out[dst] += h[src] * (dinv[src]*dinv[dst])
// lane handles columns lane + 32*j -> every atomic op covers a contiguous
// 128B segment across the wave (best case for L2 atomic units)
__global__ __launch_bounds__(256) void edge_scatter(
    const float* __restrict__ h, const long long* __restrict__ src,
    const long long* __restrict__ dst, const float* __restrict__ dinv,
    float* __restrict__ out, int E)
{
    int wid = (blockIdx.x * 256 + threadIdx.x) >> 5;
    if (wid >= E) return;
    int lane = threadIdx.x & 31;
    long long s = src[wid], d = dst[wid];
    float w = dinv[s] * dinv[d];
    const float* hs = h + (size_t)s * 128;
    float* od = out + (size_t)d * 128;
#pragma unroll
    for (int j = 0; j < 4; ++j) {
        int c = lane + j * 32;
        atomicAdd(&od[c], hs[c] * w);
    }
}

// ---------------------------------------------------------------------------
// BatchNorm statistics: stats[c] += sum, stats[128+c] += sum of squares
// thread t of each block owns channel t%128, walks rows strided by gridDim*2
// ---------------------------------------------------------------------------
__global__ __launch_bounds__(256) void bn_stats(const float* __restrict__ x,
                                                float* __restrict__ stats, int N) {
    int c = threadIdx.x & 127;
    int r0 = blockIdx.x * 2 + (threadIdx.x >> 7);
    float s = 0.f, sq = 0.f;
    for (int r = r0; r < N; r += gridDim.x * 2) {
        float v = x[(size_t)r * 128 + c];
        s += v;
        sq += v * v;
    }
    atomicAdd(&stats[c], s);
    atomicAdd(&stats[128 + c], sq);
}

__global__ void bn_finalize(const float* __restrict__ stats,
                            const float* __restrict__ gamma,
                            const float* __restrict__ beta,
                            float* __restrict__ scale, float* __restrict__ shift,
                            int N) {
    int c = threadIdx.x;
    if (c >= 128) return;
    float invN = 1.f / (float)N;
    float mean = stats[c] * invN;
    float var = stats[128 + c] * invN - mean * mean;
    float sc = gamma[c] * rsqrtf(var + BN_EPS);
    scale[c] = sc;
    shift[c] = beta[c] - mean * sc;
}

__global__ __launch_bounds__(256) void apply_bn_elu(float* __restrict__ x,
                                                    const float* __restrict__ scale,
                                                    const float* __restrict__ shift,
                                                    int n128) {
    int i = blockIdx.x * 256 + threadIdx.x;
    if (i < n128) {
        int c = i & 127;
        x[i] = elu1(x[i] * scale[c] + shift[c]);
    }
}

// ---------------------------------------------------------------------------
extern "C" void kernel_launch(void* const* d_in, const int* in_sizes, int n_in,
                              void* d_out, int out_size, void* d_ws, size_t ws_size,
                              hipStream_t stream) {
    (void)n_in; (void)out_size; (void)ws_size;

    const float* x       = (const float*)d_in[0];
    const long long* ei  = (const long long*)d_in[1];  // int64 edge_index [2,E]
    const float* W1      = (const float*)d_in[2];
    const float* b1      = (const float*)d_in[3];
    const float* g1      = (const float*)d_in[4];
    const float* be1     = (const float*)d_in[5];
    const float* W2      = (const float*)d_in[6];
    const float* b2      = (const float*)d_in[7];
    const float* g2      = (const float*)d_in[8];
    const float* be2     = (const float*)d_in[9];

    const int N = in_sizes[0] / 128;
    const int E = in_sizes[1] / 2;
    const long long* src = ei;
    const long long* dst = ei + E;

    // workspace layout
    char* ws = (char*)d_ws;
    size_t off = 0;
    auto alloc = [&](size_t bytes) {
        void* p = ws + off;
        off = (off + bytes + 255) & ~(size_t)255;
        return p;
    };
    float* dinv  = (float*)alloc((size_t)N * 4);
    float* h     = (float*)alloc((size_t)N * 128 * 4);
    float* out1  = (float*)alloc((size_t)N * 128 * 4);
    float* stats = (float*)alloc(256 * 4);
    float* scl   = (float*)alloc(128 * 4);
    float* shf   = (float*)alloc(128 * 4);
    float* outF  = (float*)d_out;

    const int n128 = N * 128;
    dim3 b256(256);
    auto cdiv = [](int a, int b) { return (a + b - 1) / b; };

    // normalization: deg = in-degree + 1; dinv = rsqrt(deg)
    zero_f32<<<cdiv(N, 256), b256, 0, stream>>>(dinv, N);
    deg_count<<<cdiv(E, 256), b256, 0, stream>>>(dst, dinv, E);
    finish_dinv<<<cdiv(N, 256), b256, 0, stream>>>(dinv, N);

    // ---- layer 1 ----
    gemm_nn_128<false><<<cdiv(N, 128), b256, 0, stream>>>(x, W1, nullptr, nullptr, h, N);
    init_out<<<cdiv(n128, 256), b256, 0, stream>>>(h, dinv, b1, out1, n128);
    edge_scatter<<<cdiv(E, 8), b256, 0, stream>>>(h, src, dst, dinv, out1, E);
    zero_f32<<<1, b256, 0, stream>>>(stats, 256);
    bn_stats<<<256, b256, 0, stream>>>(out1, stats, N);
    bn_finalize<<<1, dim3(128), 0, stream>>>(stats, g1, be1, scl, shf, N);

    // ---- layer 2 (BN1+ELU fused into GEMM A-operand read) ----
    gemm_nn_128<true><<<cdiv(N, 128), b256, 0, stream>>>(out1, W2, scl, shf, h, N);
    init_out<<<cdiv(n128, 256), b256, 0, stream>>>(h, dinv, b2, outF, n128);
    edge_scatter<<<cdiv(E, 8), b256, 0, stream>>>(h, src, dst, dinv, outF, E);
    zero_f32<<<1, b256, 0, stream>>>(stats, 256);
    bn_stats<<<256, b256, 0, stream>>>(outF, stats, N);
    bn_finalize<<<1, dim3(128), 0, stream>>>(stats, g2, be2, scl, shf, N);
    apply_bn_elu<<<cdiv(n128, 256), b256, 0, stream>>>(outF, scl, shf, n128);
}